// IterativeSequential2D_42365557408431
// MI455X (gfx1250) — compile-verified
//
#include <hip/hip_runtime.h>
#include <hip/hip_bf16.h>

typedef __attribute__((ext_vector_type(2))) float v2f;
typedef __attribute__((ext_vector_type(8))) float v8f;

#define NBLK 16
#define DSZ  256
#define BSZ  2048

// ---------------------------------------------------------------------------
// bias_full[m][e] = sum over present n (n <= m+1) of b[m][n][e]
// ---------------------------------------------------------------------------
__global__ __launch_bounds__(DSZ) void iterseq_bias_sum(
    const float* __restrict__ b, float* __restrict__ bias_full) {
  const int m = blockIdx.x;
  const int e = threadIdx.x;
  const int ncnt = (m + 2 < NBLK) ? (m + 2) : NBLK;
  float s = 0.f;
  for (int n = 0; n < ncnt; ++n)
    s += b[(m * NBLK + n) * DSZ + e];
  bias_full[m * DSZ + e] = s;
}

// ---------------------------------------------------------------------------
// Block GEMM + bias + relu using V_WMMA_F32_16X16X4_F32 (exact fp32 matmul).
//   Yout[m][bi][e] = relu( sum_{n<ncnt} sum_d Yin[n][bi][d] * W[m][n][e][d]
//                          + bias[m][e] )
// Block: 128 threads = 4 waves arranged 2x2; wave tile = 64(M) x 64(N)
// (4x4 WMMA C-tiles, 128 acc VGPRs); workgroup tile = 128 x 128.
// Grid = (B/128, D/128, NBLK).
//
// WMMA f32 16x16x4 register layouts (ISA 7.12.2):
//   A (16x4):  lane L holds row M=(L&15), K = {0,1} (L<16) or {2,3} (L>=16)
//   B (4x16):  lane L holds col N=(L&15), K = {0,1} (L<16) or {2,3} (L>=16)
//   C (16x16): vgpr j, lanes 0-15 -> M=j, lanes 16-31 -> M=j+8; N = lane&15
// ---------------------------------------------------------------------------
__global__ __launch_bounds__(128) void iterseq_gemm_relu(
    const float* __restrict__ Yin,        // [ncnt blocks of B*D] or x0 [B*D]
    const float* __restrict__ W,          // [16][16][256][256] (e-major, d inner)
    const float* __restrict__ bias_base,  // bias_base + m*bias_stride_m -> [256]
    int bias_stride_m,
    float* __restrict__ Yout,             // [16][B][256]
    int n_cnt_in,                          // >=0: fixed count; <0: min(m+2,16)
    long long in_blk_stride)               // element stride between input blocks
{
  const int m    = blockIdx.z;
  const int lane = threadIdx.x & 31;
  const int wave = threadIdx.x >> 5;
  const int wm   = wave >> 1;           // wave row in 2x2
  const int wn   = wave & 1;            // wave col in 2x2
  const int h    = lane >> 4;           // half-lane select (K-pair)
  const int l    = lane & 15;           // row/col within tile

  const int row0 = blockIdx.x * 128 + wm * 64;  // output row base (batch dim)
  const int col0 = blockIdx.y * 128 + wn * 64;  // output col base (feature dim)

  const int n_cnt = (n_cnt_in < 0) ? ((m + 2 < NBLK) ? (m + 2) : NBLK)
                                   : n_cnt_in;

  v8f acc[4][4];
  const v8f vzero = {0.f, 0.f, 0.f, 0.f, 0.f, 0.f, 0.f, 0.f};
#pragma unroll
  for (int t = 0; t < 4; ++t)
#pragma unroll
    for (int s = 0; s < 4; ++s) acc[t][s] = vzero;

  for (int n = 0; n < n_cnt; ++n) {
    const float* __restrict__ A  = Yin + (long long)n * in_blk_stride;
    const float* __restrict__ Wn = W + (long long)(m * NBLK + n) * (DSZ * DSZ);

    // Per-lane fragment base pointers (walk contiguously along K).
    const float* a_ptr[4];
#pragma unroll
    for (int t = 0; t < 4; ++t)
      a_ptr[t] = A + (long long)(row0 + t * 16 + l) * DSZ + 2 * h;
    const float* b_ptr[4];
#pragma unroll
    for (int s = 0; s < 4; ++s)
      b_ptr[s] = Wn + (long long)(col0 + s * 16 + l) * DSZ + 2 * h;

#pragma unroll 2
    for (int k = 0; k < DSZ; k += 4) {
      v2f af[4], bf[4];
#pragma unroll
      for (int t = 0; t < 4; ++t) af[t] = *(const v2f*)(a_ptr[t] + k);
#pragma unroll
      for (int s = 0; s < 4; ++s) bf[s] = *(const v2f*)(b_ptr[s] + k);
#pragma unroll
      for (int t = 0; t < 4; ++t)
#pragma unroll
        for (int s = 0; s < 4; ++s)
          acc[t][s] = __builtin_amdgcn_wmma_f32_16x16x4_f32(
              /*neg_a=*/false, af[t], /*neg_b=*/false, bf[s],
              /*c_mod=*/(short)0, acc[t][s],
              /*reuse_a=*/false, /*reuse_b=*/false);
    }
  }

  // Epilogue: bias + relu + store (C/D layout: vgpr j -> M=j+8h, N=lane&15).
  const float* __restrict__ bias = bias_base + (long long)m * bias_stride_m;
  float* __restrict__ outm = Yout + (long long)m * (BSZ * DSZ);
#pragma unroll
  for (int s = 0; s < 4; ++s) {
    const int col = col0 + s * 16 + l;
    const float bv = bias[col];
#pragma unroll
    for (int t = 0; t < 4; ++t) {
      const int rbase = row0 + t * 16 + 8 * h;
#pragma unroll
      for (int j = 0; j < 8; ++j) {
        float v = acc[t][s][j] + bv;
        v = v > 0.f ? v : 0.f;
        outm[(long long)(rbase + j) * DSZ + col] = v;
      }
    }
  }
}

// ---------------------------------------------------------------------------
// Launch: bias reduction, then 3 dependent GEMM iterations (ping-pong d_out/ws)
// ---------------------------------------------------------------------------
extern "C" void kernel_launch(void* const* d_in, const int* in_sizes, int n_in,
                              void* d_out, int out_size, void* d_ws, size_t ws_size,
                              hipStream_t stream) {
  const float* x0 = (const float*)d_in[0];  // [B, D]
  const float* W  = (const float*)d_in[1];  // [N, N, D, D]
  const float* b  = (const float*)d_in[2];  // [N, N, D]
  float* out = (float*)d_out;               // [N, B, D]

  float* ws_bias = (float*)d_ws;                 // [N, D]      (16 KB)
  float* ws_Y    = (float*)d_ws + NBLK * DSZ;    // [N, B, D]   (32 MB)

  const long long blk_stride = (long long)BSZ * DSZ;

  iterseq_bias_sum<<<NBLK, DSZ, 0, stream>>>(b, ws_bias);

  dim3 grid(BSZ / 128, DSZ / 128, NBLK);
  dim3 blk(128);

  // Iteration 1: Y = relu(x0 @ W[m,0]^T + b[m,0])           -> d_out
  iterseq_gemm_relu<<<grid, blk, 0, stream>>>(
      x0, W, b, /*bias_stride_m=*/NBLK * DSZ, out, /*n_cnt=*/1, /*stride=*/0LL);

  // Iteration 2: Y = relu(sum_n Y @ W[m,n]^T + bias_full)   -> ws_Y
  iterseq_gemm_relu<<<grid, blk, 0, stream>>>(
      out, W, ws_bias, /*bias_stride_m=*/DSZ, ws_Y, /*n_cnt=*/-1, blk_stride);

  // Iteration 3: same                                        -> d_out
  iterseq_gemm_relu<<<grid, blk, 0, stream>>>(
      ws_Y, W, ws_bias, /*bias_stride_m=*/DSZ, out, /*n_cnt=*/-1, blk_stride);
}